// GraphEncoder_57363583205747
// MI455X (gfx1250) — compile-verified
//
#include <hip/hip_runtime.h>

typedef __attribute__((ext_vector_type(2))) float v2f;
typedef __attribute__((ext_vector_type(8))) float v8f;

#define VNUM  100000
#define DIM   64
#define NEDGE 1000000
#define EPS   1e-7f

// ---------------------------------------------------------------------------
// Kernel 0: zero the ptr accumulator (d_ws) — required every launch since the
// harness does not re-poison scratch between replays.
// ---------------------------------------------------------------------------
__global__ void zero_kernel(float* __restrict__ p, int n) {
    int i = blockIdx.x * blockDim.x + threadIdx.x;
    if (i < n) p[i] = 0.0f;
}

// ---------------------------------------------------------------------------
// Kernel 1: edge gather + weighted scatter-add.
// One wave32 per edge; lane L handles dims [2L, 2L+1]. Gather is a coalesced
// float2 load of vrepr[s]; scatter is two global_atomic_add_f32 into ptr[t].
// Both tables are L2-resident (25.6 MB each << 192 MB L2), so atomics resolve
// in L2. This phase is the runtime bottleneck (64M f32 L2 atomics).
// ---------------------------------------------------------------------------
__global__ __launch_bounds__(256) void scatter_kernel(
    const int*   __restrict__ eidx,    // [2, NEDGE]
    const float* __restrict__ enorm,   // [NEDGE]
    const float* __restrict__ esgn,    // [NEDGE]
    const float* __restrict__ vrepr,   // [VNUM, DIM]
    float*       __restrict__ ptrb)    // [VNUM, DIM] accumulator
{
    const int lane = threadIdx.x & 31;
    const int wib  = threadIdx.x >> 5;                 // wave in block (0..7)
    const long long e = (long long)blockIdx.x * 8 + wib;
    if (e >= NEDGE) return;

    const int   s = eidx[e];
    const int   t = eidx[NEDGE + e];
    const float w = esgn[e] * enorm[e];

    const float2 v = ((const float2*)(vrepr + (size_t)s * DIM))[lane];
    float* dst = ptrb + (size_t)t * DIM + lane * 2;
    unsafeAtomicAdd(dst + 0, v.x * w);
    unsafeAtomicAdd(dst + 1, v.y * w);
}

// ---------------------------------------------------------------------------
// Kernel 2: dual dense heads via V_WMMA_F32_16X16X4_F32 (exact f32 math).
//
// Weights are staged ONCE per workgroup into LDS, pre-swizzled into the exact
// WMMA B-fragment layout:
//   sB[mat][nt][kstep][lane] = ( W[kb][n], W[kb+1][n] )
//   kb = 4*kstep + (lane>=16 ? 2 : 0),  n = 16*nt + (lane&15)
// so the inner loop needs ONE conflict-free ds_load_b64 per B fragment
// (32 lanes x 8B = 256B = all 64 banks) instead of 2 strided global loads.
// This cuts global B traffic from ~200 MB (per-wave reload) to ~25 MB
// (per-block), leaving the GEMM bound by the single 25.6 MB ptr read.
//
// One wave per 16-row tile of ptr; 8 accumulators (4 N-tiles x 2 heads).
// A 16x4 layout (ISA 7.12.2): lanes 0-15 hold M=lane, K=k0..k0+1; lanes 16-31
// hold K=k0+2..k0+3 -> one contiguous global_load_b64 per lane per K-step.
// C/D: VGPR r holds row (r or r+8), col = lane&15.
// ---------------------------------------------------------------------------
__global__ __launch_bounds__(256) void gemm_heads_kernel(
    const float* __restrict__ ptrb,    // [VNUM, DIM]
    const float* __restrict__ Wl,      // [DIM, DIM]
    const float* __restrict__ bl,      // [DIM]
    const float* __restrict__ Ws,      // [DIM, DIM]
    const float* __restrict__ bs,      // [DIM]
    float*       __restrict__ out)     // loc at [0, V*D), std at [V*D, 2*V*D)
{
    // [mat][nt][kstep][lane] float2 = 2*4*16*32*8B = 32 KB
    __shared__ float2 sB[2][4][16][32];

    const int tid  = threadIdx.x;
    const int lane = tid & 31;

    // ---- Stage both weight matrices into LDS, pre-swizzled (all threads) ----
    for (int i = tid; i < 2 * 4 * 16 * 32; i += 256) {
        const int ln    = i & 31;
        const int kstep = (i >> 5) & 15;
        const int nt    = (i >> 9) & 3;
        const int mat   = (i >> 11) & 1;
        const int kb    = kstep * 4 + ((ln & 16) ? 2 : 0);
        const int n     = nt * 16 + (ln & 15);
        const float* W  = mat ? Ws : Wl;
        float2 f;
        f.x = W[(size_t)kb * DIM + n];
        f.y = W[(size_t)(kb + 1) * DIM + n];
        sB[mat][nt][kstep][ln] = f;
    }
    __syncthreads();

    const int wave  = (blockIdx.x * blockDim.x + tid) >> 5;
    const int ntile = VNUM / 16;                       // 6250, exact
    if (wave >= ntile) return;                         // whole-wave exit after barrier

    const int row0  = wave * 16;
    const int m     = lane & 15;
    const int khalf = (lane & 16) ? 2 : 0;             // K sub-offset for this half-wave

    v8f accL[4] = {{0,0,0,0,0,0,0,0},{0,0,0,0,0,0,0,0},
                   {0,0,0,0,0,0,0,0},{0,0,0,0,0,0,0,0}};
    v8f accS[4] = {{0,0,0,0,0,0,0,0},{0,0,0,0,0,0,0,0},
                   {0,0,0,0,0,0,0,0},{0,0,0,0,0,0,0,0}};

    const float2* arow = (const float2*)(ptrb + (size_t)(row0 + m) * DIM);

#pragma unroll 4
    for (int kstep = 0; kstep < 16; ++kstep) {
        const int kb = kstep * 4 + khalf;
        const float2 av = arow[kb >> 1];               // K=kb, kb+1 for this lane
        v2f a; a.x = av.x; a.y = av.y;

#pragma unroll
        for (int nt = 0; nt < 4; ++nt) {
            const float2 fL = sB[0][nt][kstep][lane];  // ds_load_b64, conflict-free
            const float2 fS = sB[1][nt][kstep][lane];
            v2f bL; bL.x = fL.x; bL.y = fL.y;
            v2f bS; bS.x = fS.x; bS.y = fS.y;
            accL[nt] = __builtin_amdgcn_wmma_f32_16x16x4_f32(
                false, a, false, bL, (short)0, accL[nt], false, false);
            accS[nt] = __builtin_amdgcn_wmma_f32_16x16x4_f32(
                false, a, false, bS, (short)0, accS[nt], false, false);
        }
    }

    // Epilogue: bias, softplus(std)+EPS, coalesced stores.
    const int rbase = row0 + ((lane & 16) ? 8 : 0);
#pragma unroll
    for (int nt = 0; nt < 4; ++nt) {
        const int   n     = nt * 16 + m;
        const float biasL = bl[n];
        const float biasS = bs[n];
#pragma unroll
        for (int r = 0; r < 8; ++r) {
            const int   row = rbase + r;
            const float xl  = accL[nt][r] + biasL;
            const float xs  = accS[nt][r] + biasS;
            // stable softplus: max(x,0) + log1p(exp(-|x|))
            const float sp  = fmaxf(xs, 0.0f) + log1pf(expf(-fabsf(xs))) + EPS;
            out[(size_t)row * DIM + n] = xl;
            out[(size_t)VNUM * DIM + (size_t)row * DIM + n] = sp;
        }
    }
}

// ---------------------------------------------------------------------------
// Launch wrapper
// ---------------------------------------------------------------------------
extern "C" void kernel_launch(void* const* d_in, const int* in_sizes, int n_in,
                              void* d_out, int out_size, void* d_ws, size_t ws_size,
                              hipStream_t stream) {
    const int*   eidx  = (const int*)  d_in[0];   // [2, NEDGE] (int)
    const float* enorm = (const float*)d_in[1];   // [NEDGE]
    const float* esgn  = (const float*)d_in[2];   // [NEDGE]
    const float* vrepr = (const float*)d_in[3];   // [VNUM, DIM]
    const float* Wl    = (const float*)d_in[4];   // [DIM, DIM]
    const float* bl    = (const float*)d_in[5];   // [DIM]
    const float* Ws    = (const float*)d_in[6];   // [DIM, DIM]
    const float* bs    = (const float*)d_in[7];   // [DIM]
    float*       out   = (float*)d_out;
    float*       ptrb  = (float*)d_ws;            // V*D f32 = 25.6 MB scratch

    const int nacc = VNUM * DIM;
    zero_kernel<<<(nacc + 255) / 256, 256, 0, stream>>>(ptrb, nacc);

    scatter_kernel<<<(NEDGE + 7) / 8, 256, 0, stream>>>(eidx, enorm, esgn, vrepr, ptrb);

    const int waves = VNUM / 16;                  // 6250 row tiles
    gemm_heads_kernel<<<(waves + 7) / 8, 256, 0, stream>>>(ptrb, Wl, bl, Ws, bs, out);
}